// Coordinate_49976239456708
// MI455X (gfx1250) — compile-verified
//
#include <hip/hip_runtime.h>
#include <math.h>

typedef float v2f __attribute__((ext_vector_type(2)));
typedef float v4f __attribute__((ext_vector_type(4)));
typedef float v8f __attribute__((ext_vector_type(8)));

#define HH 64
#define WW 256
#define HWSZ (HH * WW)          // 16384 grid cells
#define NPTS 16384
#define TAU_F 2.0f
#define MIN_D 1.45f
#define MAX_D 80.0f

// ---------------------------------------------------------------------------
// Kernel 1: per-point preprocessing.
//   au = atan2(z, |xy|), av = atan2(y, x)
//   w  = exp(-tau*d) * inrange,  wd = w * d*80
// ---------------------------------------------------------------------------
__global__ void prep_points(const float* __restrict__ xyz,
                            float* __restrict__ au, float* __restrict__ av,
                            float* __restrict__ wd, float* __restrict__ wgt) {
    int i = blockIdx.x * blockDim.x + threadIdx.x;
    if (i >= NPTS) return;
    float x = xyz[3 * i + 0];
    float y = xyz[3 * i + 1];
    float z = xyz[3 * i + 2];
    float rxy = sqrtf(x * x + y * y);
    float d   = sqrtf(x * x + y * y + z * z);
    float w   = expf(-TAU_F * d);
    float d1  = d * MAX_D;
    float inr = (d1 > MIN_D && d1 < MAX_D) ? 1.0f : 0.0f;
    w *= inr;
    au[i]  = atan2f(z, rxy);
    av[i]  = atan2f(y, x);
    wd[i]  = w * d1;
    wgt[i] = w;
}

// ---------------------------------------------------------------------------
// Kernel 2: grid preprocessing from the provided angle map (1,2,H,W).
//   Packs {-2u, -2v, u^2+v^2, 0} into one float4 per grid cell so the GEMM
//   inner loop needs a single global_load_b128 per column tile. Also zeroes
//   the num/den accumulators each call (harness does not re-poison).
// ---------------------------------------------------------------------------
__global__ void prep_grid(const float* __restrict__ angle,
                          v4f* __restrict__ g4,
                          float* __restrict__ num, float* __restrict__ den) {
    int i = blockIdx.x * blockDim.x + threadIdx.x;
    if (i >= HWSZ) return;
    float u = angle[i];          // channel 0 (elevation), flat H*W
    float v = angle[HWSZ + i];   // channel 1 (azimuth)
    v4f g;
    g.x = -2.0f * u;
    g.y = -2.0f * v;
    g.z = u * u + v * v;
    g.w = 0.0f;
    g4[i] = g;
    num[i] = 0.0f;
    den[i] = 0.0f;
}

// ---------------------------------------------------------------------------
// Kernel 3: fused GEMM(K=2, via V_WMMA_F32_16X16X4_F32) + row argmin + scatter.
//   One wave32 owns a 16-point row tile. A (16x4 f32): lanes 0-15 carry
//   K=0,1 in v0,v1; lanes 16-31 carry the K=2,3 zero padding. B (4x16):
//   same K split across lane halves, column = tile*16 + (lane&15).
//   C = gsq[col] broadcast into all 8 row slots, so D = |g|^2 - 2 a.g
//   (per-row |a|^2 is argmin-invariant). 1024 column tiles per wave.
//   B is loaded unconditionally (lanes 16-31 hit the same cache lines as
//   lanes 0-15) and zeroed with cndmask -> no per-iteration EXEC branching.
// ---------------------------------------------------------------------------
__global__ void __launch_bounds__(256)
argmin_scatter(const float* __restrict__ au, const float* __restrict__ av,
               const v4f* __restrict__ g4,
               const float* __restrict__ wd, const float* __restrict__ wgt,
               float* __restrict__ num, float* __restrict__ den) {
    const int wave    = (blockIdx.x * blockDim.x + threadIdx.x) >> 5; // 0..1023
    const int lane    = threadIdx.x & 31;
    const int half    = lane >> 4;   // 0: K=0,1 lanes / rows 0-7 of C
    const int lm      = lane & 15;
    const int rowBase = wave * 16;   // 16 points per wave
    const float hmask = (half == 0) ? 1.0f : 0.0f;

    // A operand: branchless (all lanes load, K=2,3 lanes zeroed by mask).
    v2f a;
    a.x = au[rowBase + lm] * hmask;
    a.y = av[rowBase + lm] * hmask;

    float best[8];
    int   bidx[8];
#pragma unroll
    for (int m = 0; m < 8; ++m) { best[m] = 3.402823e38f; bidx[m] = 0x7fffffff; }

#pragma unroll 2
    for (int ct = 0; ct < HWSZ / 16; ++ct) {
        const int col = ct * 16 + lm;
        const v4f g = g4[col];               // single b128 load per tile
        v2f b;
        b.x = (half == 0) ? g.x : 0.0f;      // v_cndmask, no branch
        b.y = (half == 0) ? g.y : 0.0f;
        const float csq = g.z;
        v8f c = { csq, csq, csq, csq, csq, csq, csq, csq };

        // D = A(16x4) * B(4x16) + C  ->  |g|^2 - 2 a.g  per (row, col)
        v8f dmat = __builtin_amdgcn_wmma_f32_16x16x4_f32(
            /*neg_a=*/false, a, /*neg_b=*/false, b,
            /*c_mod=*/(short)0, c, /*reuse_a=*/false, /*reuse_b=*/false);

#pragma unroll
        for (int m = 0; m < 8; ++m) {
            float dv = dmat[m];
            if (dv < best[m]) { best[m] = dv; bidx[m] = col; }  // strict <: first-min
        }
    }

    // Argmin reduce across the 16 lanes of each half (xor butterfly keeps
    // lanes within their half for offsets < 16). Ties -> lowest flat index,
    // matching jnp.argmin.
#pragma unroll
    for (int m = 0; m < 8; ++m) {
        float bv = best[m];
        int   bi = bidx[m];
#pragma unroll
        for (int off = 1; off < 16; off <<= 1) {
            float ov = __shfl_xor(bv, off, 32);
            int   oi = __shfl_xor(bi, off, 32);
            if (ov < bv || (ov == bv && oi < bi)) { bv = ov; bi = oi; }
        }
        best[m] = bv;
        bidx[m] = bi;
    }

    // uv = (ids//W, ids%W) is integral -> bilinear weights degenerate to a
    // single-pixel scatter at flat index ids. Lane 0 covers rows 0-7,
    // lane 16 covers rows 8-15.
    if (lm == 0) {
#pragma unroll
        for (int m = 0; m < 8; ++m) {
            const int pt = rowBase + half * 8 + m;
            const int id = bidx[m];
            atomicAdd(&num[id], wd[pt]);
            atomicAdd(&den[id], wgt[pt]);
        }
    }
}

// ---------------------------------------------------------------------------
// Kernel 4: normalize + emit (depth_2d, valid) concatenated as float.
// ---------------------------------------------------------------------------
__global__ void finalize(const float* __restrict__ num,
                         const float* __restrict__ den,
                         float* __restrict__ out) {
    int i = blockIdx.x * blockDim.x + threadIdx.x;
    if (i >= HWSZ) return;
    float d2    = num[i] / (den[i] + 1e-8f);
    bool  valid = (d2 != 0.0f);
    float o     = (d2 - MIN_D) * (1.0f / (MAX_D - MIN_D));
    out[i]        = valid ? o : 1.0f;
    out[HWSZ + i] = valid ? 1.0f : 0.0f;
}

// ---------------------------------------------------------------------------
extern "C" void kernel_launch(void* const* d_in, const int* in_sizes, int n_in,
                              void* d_out, int out_size, void* d_ws, size_t ws_size,
                              hipStream_t stream) {
    (void)in_sizes; (void)n_in; (void)out_size; (void)ws_size;

    const float* xyz   = (const float*)d_in[0];   // (1, 16384, 3)
    const float* angle = (const float*)d_in[1];   // (1, 2, 64, 256)
    float*       out   = (float*)d_out;           // depth_2d (16384) ++ valid (16384)

    // Workspace layout (floats): 10 * 16384 = 640 KB
    float* ws  = (float*)d_ws;
    float* au  = ws + 0 * NPTS;
    float* av  = ws + 1 * NPTS;
    float* wd  = ws + 2 * NPTS;
    float* wgt = ws + 3 * NPTS;
    v4f*   g4  = (v4f*)(ws + 4 * NPTS);           // 4 * 16384 floats
    float* num = ws + 8 * NPTS;
    float* den = ws + 9 * NPTS;

    prep_points<<<NPTS / 256, 256, 0, stream>>>(xyz, au, av, wd, wgt);
    prep_grid<<<HWSZ / 256, 256, 0, stream>>>(angle, g4, num, den);
    // 1024 row tiles -> 1024 waves -> 128 blocks of 8 waves (256 threads)
    argmin_scatter<<<128, 256, 0, stream>>>(au, av, g4, wd, wgt, num, den);
    finalize<<<HWSZ / 256, 256, 0, stream>>>(num, den, out);
}